// ExplaiNN_86242943303823
// MI455X (gfx1250) — compile-verified
//
#include <hip/hip_runtime.h>
#include <hip/hip_bf16.h>

// ---------------------------------------------------------------------------
// ExplaiNN forward (eval mode), f32 end-to-end on V_WMMA_F32_16X16X4_F32.
//   B=256, L=1000, D=4, U=300, K=19, POOL=STRIDE=7, P=140, FC=100
// ---------------------------------------------------------------------------

typedef float v2f __attribute__((ext_vector_type(2)));
typedef float v8f __attribute__((ext_vector_type(8)));

#define NB   256
#define LEN  1000
#define DD   4
#define UU   300
#define KK   19
#define PP   140
#define FCN  100
#define RED  (DD*KK)          // 76  conv reduction
#define XROW (LEN*DD)         // 4000 floats per batch
#define EPSB 1e-5f

static __device__ __forceinline__ v8f wmma_f32_k4(v2f a, v2f b, v8f c) {
    return __builtin_amdgcn_wmma_f32_16x16x4_f32(
        /*neg_a=*/false, a, /*neg_b=*/false, b,
        /*c_mod=*/(short)0, c, /*reuse_a=*/false, /*reuse_b=*/false);
}

// ===========================================================================
// Kernel 1: conv(4->300,K=19) + BN1 + exp + maxpool(7,7)  -> pooled[u][b][p]
// grid = (19 unit-tiles, 256 batches), block = 256 (8 waves)
// wave w handles position-chunks {w, w+8}; chunk = 112 positions = 16 windows
// ===========================================================================
__global__ __launch_bounds__(256) void explainn_conv_pool(
    const float* __restrict__ x,        // [B][L][D]
    const float* __restrict__ w_conv,   // [U][D][K]
    const float* __restrict__ b_conv,   // [U]
    const float* __restrict__ g1, const float* __restrict__ be1,
    const float* __restrict__ m1, const float* __restrict__ v1,
    float* __restrict__ pooled)         // [U][B][P]
{
    __shared__ __align__(16) float sw[RED * 16];        // w2[r][j], r=k*4+d
    __shared__ __align__(16) float sact[8][112 * 16];   // per-wave [pos][unit]

    const int ut  = blockIdx.x;     // unit tile 0..18
    const int b   = blockIdx.y;     // batch
    const int tid = threadIdx.x;

    // ---- stage reorganized conv weights for this 16-unit tile ----
    for (int idx = tid; idx < RED * 16; idx += 256) {
        int r = idx >> 4, j = idx & 15;
        int k = r >> 2,  d = r & 3;
        int u = ut * 16 + j;
        sw[idx] = (u < UU) ? w_conv[u * RED + d * KK + k] : 0.0f;
    }
    __syncthreads();

    const int wid  = tid >> 5;
    const int lane = tid & 31;
    const int lm   = lane & 15;     // row/col within 16
    const int lh   = lane >> 4;     // lane half

    // per-lane unit params (unit = ut*16 + lm, uniform over the 8 C VGPRs)
    const int u   = ut * 16 + lm;
    const int uc  = (u < UU) ? u : (UU - 1);
    const float s1  = g1[uc] * rsqrtf(v1[uc] + EPSB);
    // fold conv bias into the BN shift: bn(conv+bias) = conv*s1 + shift
    const float sh1 = be1[uc] - m1[uc] * s1 + b_conv[uc] * s1;

    const float* xb = x + (size_t)b * XROW;

    for (int iter = 0; iter < 2; ++iter) {
        const int chunk  = wid + iter * 8;          // 0..8 valid
        const bool active = (chunk <= 8);
        if (active) {
            const int pbase = chunk * 112;
            for (int t = 0; t < 7; ++t) {
                const int p0 = pbase + t * 16;
                if (p0 >= 980) break;               // windows only need l<=979
                v8f acc = {};
                // A row = output position p0+lm ; element idx = row*4 + K
                const int abase = (p0 + lm) * 4 + 2 * lh;   // even
                #pragma unroll
                for (int s = 0; s < KK; ++s) {
                    const int r  = s * 4;
                    int ia = abase + r;                      // even
                    ia = (ia > XROW - 2) ? (XROW - 2) : ia;  // clamp (pads only)
                    v2f a = *(const v2f*)(xb + ia);
                    v2f bb;
                    bb.x = sw[(r + 2 * lh) * 16 + lm];
                    bb.y = sw[(r + 2 * lh + 1) * 16 + lm];
                    acc = wmma_f32_k4(a, bb, acc);
                }
                // BN1 + exp, transpose into LDS: act[pos_local][unit]
                #pragma unroll
                for (int v = 0; v < 8; ++v) {
                    float e = __expf(fmaf(acc[v], s1, sh1));
                    sact[wid][(t * 16 + v + 8 * lh) * 16 + lm] = e;
                }
            }
        }
        __syncthreads();   // uniform across waves (2 iterations for everyone)
        if (active) {
            const int w0 = chunk * 16;              // first global window
            #pragma unroll
            for (int i = 0; i < 8; ++i) {
                const int o  = i * 32 + lane;       // 0..255
                const int w  = o >> 4;              // local window 0..15
                const int j  = o & 15;              // unit within tile
                const int gw = w0 + w;
                if (gw < PP) {
                    const int pl = w * 7;
                    float mx = sact[wid][pl * 16 + j];
                    #pragma unroll
                    for (int q = 1; q < 7; ++q)
                        mx = fmaxf(mx, sact[wid][(pl + q) * 16 + j]);
                    const int uu2 = ut * 16 + j;
                    if (uu2 < UU)
                        pooled[((size_t)uu2 * NB + b) * PP + gw] = mx;
                }
            }
        }
        __syncthreads();
    }
}

// ===========================================================================
// Kernel 2: per-unit FC1 + BN2 + ReLU + FC2 + BN3 + ReLU + w_out accumulate
// grid = (300 units, 16 batch-tiles), block = 224 (7 waves = 7 f-tiles)
// ===========================================================================
__global__ __launch_bounds__(224) void explainn_fc(
    const float* __restrict__ pooled,   // [U][B][P]
    const float* __restrict__ w_fc1,    // [U][FC][P]
    const float* __restrict__ b_fc1,    // [U][FC]
    const float* __restrict__ g2, const float* __restrict__ be2,
    const float* __restrict__ m2, const float* __restrict__ v2,
    const float* __restrict__ w_fc2,    // [U][FC]
    const float* __restrict__ b_fc2,    // [U]
    const float* __restrict__ g3, const float* __restrict__ be3,
    const float* __restrict__ m3, const float* __restrict__ v3,
    const float* __restrict__ w_out,    // [U]
    float* __restrict__ accum)          // [B]
{
    __shared__ __align__(16) float sA[16 * PP];   // pooled[16 batches][140]
    __shared__ float zacc[16];

    const int u   = blockIdx.x;   // 0..299
    const int bt  = blockIdx.y;   // 0..15
    const int tid = threadIdx.x;

    for (int idx = tid; idx < 16 * PP; idx += 224) {
        const int row = idx / PP, p = idx - row * PP;
        sA[idx] = pooled[((size_t)u * NB + bt * 16 + row) * PP + p];
    }
    if (tid < 16) zacc[tid] = 0.0f;
    __syncthreads();

    const int ft   = tid >> 5;    // f-tile 0..6
    const int lane = tid & 31;
    const int lm   = lane & 15;
    const int lh   = lane >> 4;

    const int f  = ft * 16 + lm;              // 0..111 (valid < 100)
    const int fc = (f < FCN) ? f : (FCN - 1);

    v8f acc = {};
    const float* wrow = w_fc1 + ((size_t)u * FCN + fc) * PP;
    #pragma unroll 7
    for (int s = 0; s < 35; ++s) {
        const int r = s * 4 + 2 * lh;                 // even
        v2f a  = *(const v2f*)(sA + lm * PP + r);     // A: batch rows
        v2f bb = *(const v2f*)(wrow + r);             // B: w_fc1[u][f][p..p+1]
        acc = wmma_f32_k4(a, bb, acc);
    }

    // per-lane (u,f) params; fold b_fc1 into BN2 shift
    const int uf  = u * FCN + fc;
    const float s2  = g2[uf] * rsqrtf(v2[uf] + EPSB);
    const float sh2 = be2[uf] - m2[uf] * s2 + b_fc1[uf] * s2;
    const float wf2 = (f < FCN) ? w_fc2[uf] : 0.0f;

    // z partials per batch row m = v + 8*lh; reduce over f (lanes lm=0..15)
    #pragma unroll
    for (int v = 0; v < 8; ++v) {
        float h = fmaf(acc[v], s2, sh2);
        h = fmaxf(h, 0.0f) * wf2;
        // butterfly sum across the 16 lanes that share lh
        h += __shfl_xor(h, 1, 32);
        h += __shfl_xor(h, 2, 32);
        h += __shfl_xor(h, 4, 32);
        h += __shfl_xor(h, 8, 32);
        if (lm == 0) atomicAdd(&zacc[v + 8 * lh], h);
    }
    __syncthreads();

    if (tid < 16) {
        const float z  = zacc[tid] + b_fc2[u];
        const float s3 = g3[u] * rsqrtf(v3[u] + EPSB);
        float zz = fmaf(z - m3[u], s3, be3[u]);
        zz = fmaxf(zz, 0.0f) * w_out[u];
        atomicAdd(&accum[bt * 16 + tid], zz);
    }
}

// ===========================================================================
// Kernel 3: sigmoid(accum + b_out) -> out[B]
// ===========================================================================
__global__ __launch_bounds__(256) void explainn_out(
    const float* __restrict__ accum, const float* __restrict__ b_out,
    float* __restrict__ out)
{
    const int b = threadIdx.x;
    if (b < NB) {
        const float t = accum[b] + b_out[0];
        out[b] = 1.0f / (1.0f + __expf(-t));
    }
}

// ===========================================================================
extern "C" void kernel_launch(void* const* d_in, const int* in_sizes, int n_in,
                              void* d_out, int out_size, void* d_ws, size_t ws_size,
                              hipStream_t stream) {
    const float* x      = (const float*)d_in[0];
    const float* w_conv = (const float*)d_in[1];
    const float* b_conv = (const float*)d_in[2];
    const float* g1  = (const float*)d_in[3];
    const float* be1 = (const float*)d_in[4];
    const float* m1  = (const float*)d_in[5];
    const float* v1  = (const float*)d_in[6];
    const float* w_fc1 = (const float*)d_in[7];
    const float* b_fc1 = (const float*)d_in[8];
    const float* g2  = (const float*)d_in[9];
    const float* be2 = (const float*)d_in[10];
    const float* m2  = (const float*)d_in[11];
    const float* v2  = (const float*)d_in[12];
    const float* w_fc2 = (const float*)d_in[13];
    const float* b_fc2 = (const float*)d_in[14];
    const float* g3  = (const float*)d_in[15];
    const float* be3 = (const float*)d_in[16];
    const float* m3  = (const float*)d_in[17];
    const float* v3  = (const float*)d_in[18];
    const float* w_out = (const float*)d_in[19];
    const float* b_out = (const float*)d_in[20];

    float* pooled = (float*)d_ws;                         // U*B*P f32 = 43 MB
    float* accum  = pooled + (size_t)UU * NB * PP;        // B f32

    hipMemsetAsync(accum, 0, NB * sizeof(float), stream);

    explainn_conv_pool<<<dim3(19, NB), 256, 0, stream>>>(
        x, w_conv, b_conv, g1, be1, m1, v1, pooled);

    explainn_fc<<<dim3(UU, 16), 224, 0, stream>>>(
        pooled, w_fc1, b_fc1, g2, be2, m2, v2,
        w_fc2, b_fc2, g3, be3, m3, v3, w_out, accum);

    explainn_out<<<1, 256, 0, stream>>>(accum, b_out, (float*)d_out);
}